// DynamicGraphConvolution_85727547228211
// MI455X (gfx1250) — compile-verified
//
#include <hip/hip_runtime.h>
#include <hip/hip_bf16.h>

// GCNII-style dynamic graph convolution for MI455X (gfx1250, wave32).
//
// Algebraic refactor:  out = relu(BETA*(S@W) + (1-BETA)*S) = relu(S @ W')
// with  W' = BETA*W + (1-BETA)*I  (precomputed, stored TRANSPOSED so B-frag
// K-pairs are contiguous 8-byte loads).
//
// Phase 0: Wt[n][k] = BETA*W[k][n] + (k==n)*(1-BETA)       (16 KB, one-time)
// Phase 1: support = ALPHA * features0                      (init accumulator)
// Phase 2: support[dst] += (1-ALPHA)*val * features[src]    (f32 atomics, L2)
// Phase 3: out = relu(support @ W')                         (WMMA f32 16x16x4)

#define N_NODES 100000
#define N_EDGES 1000000
#define FDIM 64
#define ALPHA 0.1f
#define BETA 0.5f

typedef float v2f __attribute__((ext_vector_type(2)));
typedef float v8f __attribute__((ext_vector_type(8)));

// ---------------------------------------------------------------------------
// Phase 0: W' = BETA*W + (1-BETA)*I, stored transposed (Wt[n*64+k]).
// ---------------------------------------------------------------------------
__global__ __launch_bounds__(256) void gcn_prep_w(const float* __restrict__ W,
                                                  float* __restrict__ Wt) {
  int idx = blockIdx.x * blockDim.x + threadIdx.x;  // 0..4095
  if (idx < FDIM * FDIM) {
    int k = idx >> 6;
    int n = idx & 63;
    float v = BETA * W[k * FDIM + n] + ((k == n) ? (1.0f - BETA) : 0.0f);
    Wt[n * FDIM + k] = v;
  }
}

// ---------------------------------------------------------------------------
// Phase 1: support = ALPHA * features0   (vectorized float4, one pass)
// ---------------------------------------------------------------------------
__global__ __launch_bounds__(256) void gcn_init_support(
    const float4* __restrict__ f0, float4* __restrict__ support, int n4) {
  int i = blockIdx.x * blockDim.x + threadIdx.x;
  if (i < n4) {
    float4 v = f0[i];
    v.x *= ALPHA; v.y *= ALPHA; v.z *= ALPHA; v.w *= ALPHA;
    support[i] = v;
  }
}

// ---------------------------------------------------------------------------
// Phase 2: edge scatter. One wave (32 lanes) per edge; each lane handles a
// float2 slice of the 64-wide feature row. Gather/scatter are L2-resident
// (features + support = 51 MB << 192 MB L2); native global_atomic_add_f32.
// ---------------------------------------------------------------------------
__global__ __launch_bounds__(256) void gcn_edge_scatter(
    const float* __restrict__ features,
    const int* __restrict__ edge_src,
    const int* __restrict__ edge_dst,
    const float* __restrict__ edge_vals,
    float* __restrict__ support, int n_edges) {
  int wave_in_blk = threadIdx.x >> 5;
  int lane = threadIdx.x & 31;
  int e = blockIdx.x * 8 + wave_in_blk;
  if (e >= n_edges) return;

  int src = edge_src[e];
  int dst = edge_dst[e];
  float w = edge_vals[e] * (1.0f - ALPHA);

  const float2* frow =
      (const float2*)(features + (size_t)src * FDIM) + lane;  // 8B per lane
  float2 m = *frow;
  m.x *= w;
  m.y *= w;

  float* drow = support + (size_t)dst * FDIM + 2 * lane;
  __hip_atomic_fetch_add(drow + 0, m.x, __ATOMIC_RELAXED,
                         __HIP_MEMORY_SCOPE_AGENT);
  __hip_atomic_fetch_add(drow + 1, m.y, __ATOMIC_RELAXED,
                         __HIP_MEMORY_SCOPE_AGENT);
}

// ---------------------------------------------------------------------------
// Phase 3: out = relu(support @ W')
// One wave per 32-row tile (two 16-row M tiles); 2 A-frags x 4 B-frags = 8
// V_WMMA_F32_16X16X4_F32 per k-step, 16 k-steps. All fragment loads are b64.
//
// A (16x4 f32, 2 VGPRs): lanes 0-15  -> M=lane,    K = k0+{0,1}
//                        lanes 16-31 -> M=lane-16, K = k0+{2,3}
// B (4x16 f32, 2 VGPRs): lanes 0-15  -> N=lane,    K = k0+{0,1}
//                        lanes 16-31 -> N=lane-16, K = k0+{2,3}
//   (Wt is transposed, so (K,K+1) at fixed N is one contiguous float2.)
// C/D (16x16 f32, 8 VGPRs): VGPR v -> M = v + 8*(lane>=16), N = lane&15
// ---------------------------------------------------------------------------
__global__ __launch_bounds__(256) void gcn_gemm_relu(
    const float* __restrict__ support,
    const float* __restrict__ Wt,
    float* __restrict__ out, int nrows) {
  int gwave = (blockIdx.x * blockDim.x + threadIdx.x) >> 5;
  int lane = threadIdx.x & 31;
  int row0 = gwave * 32;
  if (row0 >= nrows) return;

  int half = lane >> 4;   // 0: lanes 0-15, 1: lanes 16-31
  int lidx = lane & 15;

  v8f acc[8] = {};  // [mt*4 + nt]

  // float2-typed fragment base pointers; element index k2 = k0/2 walks K.
  const v2f* arow0 =
      (const v2f*)(support + (size_t)(row0 + lidx) * FDIM) + half;
  const v2f* arow1 =
      (const v2f*)(support + (size_t)(row0 + 16 + lidx) * FDIM) + half;
  const v2f* brow0 = (const v2f*)(Wt + (size_t)(0 * 16 + lidx) * FDIM) + half;
  const v2f* brow1 = (const v2f*)(Wt + (size_t)(1 * 16 + lidx) * FDIM) + half;
  const v2f* brow2 = (const v2f*)(Wt + (size_t)(2 * 16 + lidx) * FDIM) + half;
  const v2f* brow3 = (const v2f*)(Wt + (size_t)(3 * 16 + lidx) * FDIM) + half;

#pragma unroll 4
  for (int k0 = 0; k0 < FDIM; k0 += 4) {
    int k2 = k0 >> 1;
    v2f a0 = arow0[k2];
    v2f a1 = arow1[k2];
    v2f b0 = brow0[k2];
    v2f b1 = brow1[k2];
    v2f b2 = brow2[k2];
    v2f b3 = brow3[k2];

    acc[0] = __builtin_amdgcn_wmma_f32_16x16x4_f32(false, a0, false, b0,
                                                   (short)0, acc[0], false, false);
    acc[1] = __builtin_amdgcn_wmma_f32_16x16x4_f32(false, a0, false, b1,
                                                   (short)0, acc[1], false, false);
    acc[2] = __builtin_amdgcn_wmma_f32_16x16x4_f32(false, a0, false, b2,
                                                   (short)0, acc[2], false, false);
    acc[3] = __builtin_amdgcn_wmma_f32_16x16x4_f32(false, a0, false, b3,
                                                   (short)0, acc[3], false, false);
    acc[4] = __builtin_amdgcn_wmma_f32_16x16x4_f32(false, a1, false, b0,
                                                   (short)0, acc[4], false, false);
    acc[5] = __builtin_amdgcn_wmma_f32_16x16x4_f32(false, a1, false, b1,
                                                   (short)0, acc[5], false, false);
    acc[6] = __builtin_amdgcn_wmma_f32_16x16x4_f32(false, a1, false, b2,
                                                   (short)0, acc[6], false, false);
    acc[7] = __builtin_amdgcn_wmma_f32_16x16x4_f32(false, a1, false, b3,
                                                   (short)0, acc[7], false, false);
  }

  // Epilogue: pure relu + store (skip connection already folded into W').
#pragma unroll
  for (int mt = 0; mt < 2; ++mt) {
#pragma unroll
    for (int nt = 0; nt < 4; ++nt) {
      const v8f c = acc[mt * 4 + nt];
#pragma unroll
      for (int v = 0; v < 8; ++v) {
        int r = row0 + mt * 16 + v + 8 * half;
        int cidx = nt * 16 + lidx;
        out[(size_t)r * FDIM + cidx] = fmaxf(c[v], 0.0f);
      }
    }
  }
}

// ---------------------------------------------------------------------------
extern "C" void kernel_launch(void* const* d_in, const int* in_sizes, int n_in,
                              void* d_out, int out_size, void* d_ws,
                              size_t ws_size, hipStream_t stream) {
  const float* features  = (const float*)d_in[0];
  const float* features0 = (const float*)d_in[1];
  const int*   edge_src  = (const int*)d_in[2];
  const int*   edge_dst  = (const int*)d_in[3];
  const float* edge_vals = (const float*)d_in[4];
  const float* W         = (const float*)d_in[5];
  float* out = (float*)d_out;

  float* support = (float*)d_ws;                       // 25.6 MB scratch
  float* Wt      = (float*)d_ws + (size_t)N_NODES * FDIM;  // +16 KB scratch

  // Phase 0: W' = BETA*W + (1-BETA)*I, transposed.
  gcn_prep_w<<<(FDIM * FDIM + 255) / 256, 256, 0, stream>>>(W, Wt);

  // Phase 1: support = ALPHA * features0
  int n4 = (N_NODES * FDIM) / 4;  // 1,600,000 float4 elements
  gcn_init_support<<<(n4 + 255) / 256, 256, 0, stream>>>(
      (const float4*)features0, (float4*)support, n4);

  // Phase 2: scatter edges (8 edges per 256-thread block, one wave each)
  int nblk_e = (N_EDGES + 7) / 8;
  gcn_edge_scatter<<<nblk_e, 256, 0, stream>>>(features, edge_src, edge_dst,
                                               edge_vals, support, N_EDGES);

  // Phase 3: WMMA GEMM + ReLU (one wave per 32-row tile, 8 waves per block)
  int ntiles = (N_NODES + 31) / 32;       // 3125
  int nblk_g = (ntiles + 7) / 8;          // 391
  gcn_gemm_relu<<<nblk_g, 256, 0, stream>>>(support, Wt, out, N_NODES);
}